// PPolyIntSoftmax_29308856828453
// MI455X (gfx1250) — compile-verified
//
#include <hip/hip_runtime.h>
#include <math.h>
#include <stdint.h>

#define ROW_LEN 512
#define WPB 8                 // waves per block == rows per group
#define THREADS (WPB * 32)
#define NSEG 16

// Issue 4 async b128 copies: one 2KB row (wave-cooperative) global -> LDS.
// Tracked on ASYNCcnt; completes in order with other async ops of this wave.
__device__ __forceinline__ void async_row_load(uint32_t lds_byte,
                                               const float* grow, int lane) {
#pragma unroll
  for (int j = 0; j < 4; ++j) {
    const void* g = (const void*)(grow + lane * 4 + j * 128);
    uint32_t l = lds_byte + (uint32_t)(lane * 16 + j * 512);
    asm volatile("global_load_async_to_lds_b128 %0, %1, off"
                 :: "v"(l), "v"(g) : "memory");
  }
}

__global__ __launch_bounds__(THREADS, 1) void ppoly_int_softmax(
    const float* __restrict__ x, const float* __restrict__ sf,
    const float* __restrict__ lo_bounds, const float* __restrict__ coeffs,
    float* __restrict__ out, int nrows, int out_n) {
  __shared__ __align__(16) float rowbuf[2][WPB][ROW_LEN];  // 32 KB double buffer
  __shared__ float4 ctab[NSEG];       // (a0, a1, a2, pad) per segment
  __shared__ float s_lo[NSEG + 1];    // lo_bounds + +inf sentinel

  const int tid = threadIdx.x;
  const int wave = tid >> 5;
  const int lane = tid & 31;

  if (tid < NSEG) {
    ctab[tid] = make_float4(coeffs[3 * tid], coeffs[3 * tid + 1],
                            coeffs[3 * tid + 2], 0.0f);
    s_lo[tid] = lo_bounds[tid];
    if (tid == 0) s_lo[NSEG] = INFINITY;
  }
  __syncthreads();

  const float s = sf[0];
  const int ngroups = (nrows + WPB - 1) / WPB;
  const int stride = gridDim.x;

  // Prime the pipeline: prefetch this block's first row into buffer 0.
  const int g0 = blockIdx.x;
  if (g0 < ngroups) {
    int row = g0 * WPB + wave;
    if (row < nrows)
      async_row_load((uint32_t)(uintptr_t)&rowbuf[0][wave][0],
                     x + (size_t)row * ROW_LEN, lane);
  }

  int buf = 0;
  for (int g = g0; g < ngroups; g += stride) {
    // Prefetch next group's row into the other buffer, then wait for current.
    int gn = g + stride;
    bool next_issued = false;
    if (gn < ngroups) {
      int rown = gn * WPB + wave;
      if (rown < nrows) {
        async_row_load((uint32_t)(uintptr_t)&rowbuf[buf ^ 1][wave][0],
                       x + (size_t)rown * ROW_LEN, lane);
        next_issued = true;
      }
    }
    // In-order completion: <=4 outstanding means the current row's 4 are done.
    if (next_issued) asm volatile("s_wait_asynccnt 4" ::: "memory");
    else             asm volatile("s_wait_asynccnt 0" ::: "memory");

    const int row = g * WPB + wave;
    if (row < nrows) {
      const float* rb = &rowbuf[buf][wave][0];

      // Pass 1: x_int = floor(x/s) (IEEE div, matches XLA), row max.
      float xa[16];
      float lmax = -INFINITY;
#pragma unroll
      for (int j = 0; j < 4; ++j) {
        float4 v = *(const float4*)(rb + lane * 4 + j * 128);
        float a0 = floorf(v.x / s);
        float a1 = floorf(v.y / s);
        float a2 = floorf(v.z / s);
        float a3 = floorf(v.w / s);
        xa[4 * j + 0] = a0; xa[4 * j + 1] = a1;
        xa[4 * j + 2] = a2; xa[4 * j + 3] = a3;
        lmax = fmaxf(lmax, fmaxf(fmaxf(a0, a1), fmaxf(a2, a3)));
      }
#pragma unroll
      for (int off = 16; off >= 1; off >>= 1)
        lmax = fmaxf(lmax, __shfl_xor(lmax, off, 32));
      const float shift = 127.0f - lmax;  // exact: small integers

      // Pass 2: segment lookup + quadratic Horner + floor(/2^15), row sum.
      float ev[16];
      float lsum = 0.0f;
#pragma unroll
      for (int k = 0; k < 16; ++k) {
        float xi = xa[k] + shift;
        // affine guess (16/255 per integer step), provably within {j-1, j}
        int gg = (int)floorf((xi + 128.0f) * 0.06274509803921568627f);
        gg = gg < 0 ? 0 : (gg > (NSEG - 1) ? (NSEG - 1) : gg);
        gg += (xi >= s_lo[gg + 1]) ? 1 : 0;   // single upward correction
        float4 c = ctab[gg];
        // Horner with UNFUSED mul/add (mirror XLA; defeat -ffp-contract=fast)
        float r = __fadd_rn(__fmul_rn(c.z, xi), c.y);
        r = __fadd_rn(__fmul_rn(r, xi), c.x);
        r = fmaxf(r, 0.0f);
        r = floorf(r * 3.0517578125e-05f);    // / 2^15 (exact scaling)
        ev[k] = r;
        lsum += r;
      }
#pragma unroll
      for (int off = 16; off >= 1; off >>= 1)
        lsum += __shfl_xor(lsum, off, 32);    // exact: integers, < 2^24
      const float esum = fmaxf(lsum, 1.0f);
      const float factor = floorf(4294967296.0f / esum);

      // softmax_int = floor((e*factor)/2^25); output = softmax_int * 2^-7
      float* orow = out + (size_t)row * ROW_LEN;
#pragma unroll
      for (int j = 0; j < 4; ++j) {
        float4 o;
        o.x = floorf(__fmul_rn(ev[4 * j + 0], factor) *
                     2.9802322387695312e-08f) * 0.0078125f;
        o.y = floorf(__fmul_rn(ev[4 * j + 1], factor) *
                     2.9802322387695312e-08f) * 0.0078125f;
        o.z = floorf(__fmul_rn(ev[4 * j + 2], factor) *
                     2.9802322387695312e-08f) * 0.0078125f;
        o.w = floorf(__fmul_rn(ev[4 * j + 3], factor) *
                     2.9802322387695312e-08f) * 0.0078125f;
        *(float4*)(orow + lane * 4 + j * 128) = o;
      }
    }
    buf ^= 1;
  }

  // Second tuple output: s_out = 2 / 2^8
  if (blockIdx.x == 0 && tid == 0) out[out_n - 1] = 0.0078125f;
}

extern "C" void kernel_launch(void* const* d_in, const int* in_sizes, int n_in,
                              void* d_out, int out_size, void* d_ws,
                              size_t ws_size, hipStream_t stream) {
  const float* x  = (const float*)d_in[0];
  const float* sf = (const float*)d_in[1];
  const float* lo = (const float*)d_in[2];
  const float* cf = (const float*)d_in[3];
  float* out = (float*)d_out;

  const int n = in_sizes[0];
  const int nrows = n / ROW_LEN;                       // 49152
  const int ngroups = (nrows + WPB - 1) / WPB;         // 6144
  int blocks = ngroups < 1536 ? ngroups : 1536;        // ~4 groups/block => DB pipeline
  if (blocks < 1) blocks = 1;

  ppoly_int_softmax<<<blocks, THREADS, 0, stream>>>(x, sf, lo, cf, out, nrows,
                                                    out_size);
}